// CVAEWithTrajectoryOptimization_76802605187697
// MI455X (gfx1250) — compile-verified
//
#include <hip/hip_runtime.h>
#include <hip/hip_bf16.h>

typedef float v2f __attribute__((ext_vector_type(2)));
typedef float v8f __attribute__((ext_vector_type(8)));

#define BQ      32      // batch
#define HA      112     // H*A
#define NFLAT   3584    // B*H*A
#define SZF     576     // z feature dim
#define K1      688     // SZF + HA
#define DH      512     // hidden dim
#define DAMPING_C 0.1f
#define STEP_C    0.1f
#define OFFSET_C  1000.0f

static __device__ __forceinline__ v8f wmma4(v2f a, v2f b, v8f c) {
  // D(16x16,f32) = A(16x4,f32) * B(4x16,f32) + C
  return __builtin_amdgcn_wmma_f32_16x16x4_f32(false, a, false, b, (short)0, c,
                                               false, false);
}

static __device__ __forceinline__ float eluf(float x) {
  return x > 0.f ? x : (__expf(x) - 1.f);
}
static __device__ __forceinline__ float clamp1(float x) {
  return fminf(fmaxf(x, -1.f), 1.f);
}

#define V8_ZERO {0.f,0.f,0.f,0.f,0.f,0.f,0.f,0.f}

// ---------------------------------------------------------------------------
// init: flat = init_actions ; xbuf = [ z | clip(init_actions) ]  (32 x 688)
// ---------------------------------------------------------------------------
__global__ void k_init(const float* __restrict__ init_actions,
                       const float* __restrict__ z,
                       float* __restrict__ flat, float* __restrict__ xbuf) {
  const int i = blockIdx.x * blockDim.x + threadIdx.x;   // 0..22015
  if (i >= BQ * K1) return;
  const int b = i / K1, c = i % K1;
  if (c < SZF) {
    xbuf[i] = z[b * SZF + c];
  } else {
    const float v = init_actions[b * HA + (c - SZF)];
    flat[b * HA + (c - SZF)] = v;
    xbuf[i] = clamp1(v);
  }
}

// ---------------------------------------------------------------------------
// fwd1: h1 = elu(xbuf @ W1 + b1).  Branch-free contiguous GEMM (pitch 688).
// grid 8 x 128 (4 waves). wave -> one N-tile, both M-tiles (B shared).
// stores h1 [32x512] and h1T [512x32].
// ---------------------------------------------------------------------------
__global__ void k_fwd1(const float* __restrict__ xbuf,
                       const float* __restrict__ W1, const float* __restrict__ b1,
                       float* __restrict__ h1, float* __restrict__ h1T) {
  const int lane  = threadIdx.x & 31;
  const int wave  = threadIdx.x >> 5;
  const int nTile = blockIdx.x * 4 + wave;       // 0..31
  const int nl    = lane & 15;
  const int khalf = lane >> 4;                   // 0/1
  const int n     = nTile * 16 + nl;

  const float* xr0 = xbuf + nl * K1 + 2 * khalf;          // batch rows 0..15
  const float* xr1 = xbuf + (nl + 16) * K1 + 2 * khalf;   // batch rows 16..31
  const float* wp  = W1 + 2 * khalf * DH + n;

  v8f c0 = V8_ZERO;
  v8f c1 = V8_ZERO;
#pragma unroll 4
  for (int k0 = 0; k0 < K1; k0 += 4) {
    v2f b;
    b.x = wp[k0 * DH];
    b.y = wp[k0 * DH + DH];
    v2f a0 = *(const v2f*)(xr0 + k0);
    v2f a1 = *(const v2f*)(xr1 + k0);
    c0 = wmma4(a0, b, c0);
    c1 = wmma4(a1, b, c1);
  }
  const float bias = b1[n];
#pragma unroll
  for (int v = 0; v < 8; ++v) {
    const int mr = v + 8 * khalf;
    const float h0  = eluf(c0[v] + bias);
    const float h16 = eluf(c1[v] + bias);
    h1[mr * DH + n]        = h0;
    h1[(mr + 16) * DH + n] = h16;
    h1T[n * BQ + mr]       = h0;
    h1T[n * BQ + mr + 16]  = h16;
  }
}

// ---------------------------------------------------------------------------
// fwd2: h2 = elu(h1 @ W2 + b2); dpre2T[j][b] = -(1/32)*W3[j]*elu'(pre2)
// (layer-3 backward folded into the epilogue). grid 8 x 128.
// ---------------------------------------------------------------------------
__global__ void k_fwd2(const float* __restrict__ h1, const float* __restrict__ W2,
                       const float* __restrict__ b2, const float* __restrict__ W3,
                       float* __restrict__ h2, float* __restrict__ dpre2T) {
  const int lane  = threadIdx.x & 31;
  const int wave  = threadIdx.x >> 5;
  const int nTile = blockIdx.x * 4 + wave;
  const int nl    = lane & 15;
  const int khalf = lane >> 4;
  const int n     = nTile * 16 + nl;

  const float* ar0 = h1 + nl * DH + 2 * khalf;
  const float* ar1 = h1 + (nl + 16) * DH + 2 * khalf;
  const float* wp  = W2 + 2 * khalf * DH + n;

  v8f c0 = V8_ZERO;
  v8f c1 = V8_ZERO;
#pragma unroll 4
  for (int k0 = 0; k0 < DH; k0 += 4) {
    v2f b;
    b.x = wp[k0 * DH];
    b.y = wp[k0 * DH + DH];
    v2f a0 = *(const v2f*)(ar0 + k0);
    v2f a1 = *(const v2f*)(ar1 + k0);
    c0 = wmma4(a0, b, c0);
    c1 = wmma4(a1, b, c1);
  }
  const float bias = b2[n];
  const float w3n  = W3[n];
  const float sc   = -1.f / (float)BQ;
#pragma unroll
  for (int v = 0; v < 8; ++v) {
    const int mr = v + 8 * khalf;
    const float h0  = eluf(c0[v] + bias);
    const float h16 = eluf(c1[v] + bias);
    h2[mr * DH + n]        = h0;
    h2[(mr + 16) * DH + n] = h16;
    dpre2T[n * BQ + mr]      = sc * w3n * (h0  > 0.f ? 1.f : h0  + 1.f);
    dpre2T[n * BQ + mr + 16] = sc * w3n * (h16 > 0.f ? 1.f : h16 + 1.f);
  }
}

// ---------------------------------------------------------------------------
// err: e = mean_b( 1000 - nan_to_num(h2[b,:] @ W3 + b3) ). one block of 512.
// ---------------------------------------------------------------------------
__global__ void k_err(const float* __restrict__ h2, const float* __restrict__ W3,
                      const float* __restrict__ b3, float* __restrict__ e_out) {
  __shared__ float red[512];
  __shared__ float eb[BQ];
  const int tid = threadIdx.x;
  const int b = tid >> 4, t = tid & 15;
  float s = 0.f;
  for (int j = t; j < DH; j += 16) s += h2[b * DH + j] * W3[j];
  red[tid] = s;
  __syncthreads();
  for (int off = 8; off > 0; off >>= 1) {
    if (t < off) red[tid] += red[tid + off];
    __syncthreads();
  }
  if (t == 0) {
    float r = red[tid] + b3[0];
    if (isnan(r))       r = 0.f;
    else if (isinf(r))  r = (r > 0.f) ? 3.402823466e38f : -3.402823466e38f;
    eb[b] = OFFSET_C - r;
  }
  __syncthreads();
  if (tid == 0) {
    float acc = 0.f;
    for (int i = 0; i < BQ; ++i) acc += eb[i];
    e_out[0] = acc / (float)BQ;
  }
}

// ---------------------------------------------------------------------------
// bwd1 (transposed): dpre1T[i][b] = (W2 @ dpre2T)[i][b] * elu'(h1T[i][b])
// A = W2 row-major, B = dpre2T row-major: all loads coalesced. grid 8 x 128.
// ---------------------------------------------------------------------------
__global__ void k_bwd1(const float* __restrict__ W2, const float* __restrict__ dpre2T,
                       const float* __restrict__ h1T, float* __restrict__ dpre1T) {
  const int lane  = threadIdx.x & 31;
  const int wave  = threadIdx.x >> 5;
  const int mTile = blockIdx.x * 4 + wave;   // 0..31 over i
  const int m0    = mTile * 16;
  const int nl    = lane & 15;
  const int khalf = lane >> 4;

  const float* wp = W2 + (m0 + nl) * DH + 2 * khalf;
  const float* dp = dpre2T + 2 * khalf * BQ + nl;

  v8f c0 = V8_ZERO;
  v8f c1 = V8_ZERO;
#pragma unroll 4
  for (int k0 = 0; k0 < DH; k0 += 4) {
    v2f a = *(const v2f*)(wp + k0);
    v2f bb0, bb1;
    bb0.x = dp[k0 * BQ];
    bb0.y = dp[k0 * BQ + BQ];
    bb1.x = dp[k0 * BQ + 16];
    bb1.y = dp[k0 * BQ + BQ + 16];
    c0 = wmma4(a, bb0, c0);   // batch cols 0..15
    c1 = wmma4(a, bb1, c1);   // batch cols 16..31
  }
#pragma unroll
  for (int v = 0; v < 8; ++v) {
    const int mi = m0 + v + 8 * khalf;
    const float hA = h1T[mi * BQ + nl];
    const float hB = h1T[mi * BQ + 16 + nl];
    dpre1T[mi * BQ + nl]      = c0[v] * (hA > 0.f ? 1.f : hA + 1.f);
    dpre1T[mi * BQ + 16 + nl] = c1[v] * (hB > 0.f ? 1.f : hB + 1.f);
  }
}

// ---------------------------------------------------------------------------
// bwd2 + Sherman-Morrison update, fused.
// One block (7 waves, 224 threads) covers the whole 112x32 gradient:
//   gT = (W1[576:,:] @ dpre1T) * clip_mask(flat)
//   ||g||^2 via block reduce;  flat += -STEP*e/(DAMPING+||g||^2) * g
// Also rewrites the clamped acts region of xbuf for the next iteration and
// writes d_out on the final iteration.
// ---------------------------------------------------------------------------
__global__ void k_bwd2_update(const float* __restrict__ W1,
                              const float* __restrict__ dpre1T,
                              float* __restrict__ flat,
                              const float* __restrict__ e_ptr,
                              float* __restrict__ xbuf,
                              float* __restrict__ out, int write_out) {
  __shared__ float red[224];
  const int lane  = threadIdx.x & 31;
  const int wave  = threadIdx.x >> 5;      // 0..6 -> M-tile over a
  const int m0    = wave * 16;
  const int nl    = lane & 15;
  const int khalf = lane >> 4;

  const float* wp = W1 + (SZF + m0 + nl) * DH + 2 * khalf;
  const float* dp = dpre1T + 2 * khalf * BQ + nl;

  v8f c0 = V8_ZERO;
  v8f c1 = V8_ZERO;
#pragma unroll 4
  for (int k0 = 0; k0 < DH; k0 += 4) {
    v2f a = *(const v2f*)(wp + k0);
    v2f bb0, bb1;
    bb0.x = dp[k0 * BQ];
    bb0.y = dp[k0 * BQ + BQ];
    bb1.x = dp[k0 * BQ + 16];
    bb1.y = dp[k0 * BQ + BQ + 16];
    c0 = wmma4(a, bb0, c0);
    c1 = wmma4(a, bb1, c1);
  }

  float f0[8], f1[8], g0[8], g1[8];
  float gg = 0.f;
#pragma unroll
  for (int v = 0; v < 8; ++v) {
    const int ai = m0 + v + 8 * khalf;        // 0..111
    f0[v] = flat[nl * HA + ai];
    f1[v] = flat[(16 + nl) * HA + ai];
    g0[v] = (f0[v] >= -1.f && f0[v] <= 1.f) ? c0[v] : 0.f;
    g1[v] = (f1[v] >= -1.f && f1[v] <= 1.f) ? c1[v] : 0.f;
    gg += g0[v] * g0[v] + g1[v] * g1[v];
  }
  red[threadIdx.x] = gg;
  __syncthreads();
  for (int off = 128; off > 0; off >>= 1) {
    if (threadIdx.x < off && threadIdx.x + off < 224)
      red[threadIdx.x] += red[threadIdx.x + off];
    __syncthreads();
  }
  const float scale = -STEP_C * e_ptr[0] / (DAMPING_C + red[0]);

#pragma unroll
  for (int v = 0; v < 8; ++v) {
    const int ai = m0 + v + 8 * khalf;
    const float nf0 = f0[v] + scale * g0[v];
    const float nf1 = f1[v] + scale * g1[v];
    flat[nl * HA + ai]        = nf0;
    flat[(16 + nl) * HA + ai] = nf1;
    xbuf[nl * K1 + SZF + ai]        = clamp1(nf0);
    xbuf[(16 + nl) * K1 + SZF + ai] = clamp1(nf1);
    if (write_out) {
      out[nl * HA + ai]        = nf0;
      out[(16 + nl) * HA + ai] = nf1;
    }
  }
}

// ---------------------------------------------------------------------------
extern "C" void kernel_launch(void* const* d_in, const int* in_sizes, int n_in,
                              void* d_out, int out_size, void* d_ws, size_t ws_size,
                              hipStream_t stream) {
  (void)in_sizes; (void)n_in; (void)out_size; (void)ws_size;
  const float* init_actions = (const float*)d_in[0];
  const float* z  = (const float*)d_in[1];
  const float* W1 = (const float*)d_in[2];
  const float* b1 = (const float*)d_in[3];
  const float* W2 = (const float*)d_in[4];
  const float* b2 = (const float*)d_in[5];
  const float* W3 = (const float*)d_in[6];
  const float* b3 = (const float*)d_in[7];
  float* out = (float*)d_out;

  float* ws     = (float*)d_ws;
  float* flat   = ws;                 // 3584
  float* e      = ws + 3584;          // 1 (padded to 32)
  float* xbuf   = ws + 3616;          // 32x688 = 22016
  float* h1     = ws + 25632;         // 32x512
  float* h1T    = ws + 42016;         // 512x32
  float* h2     = ws + 58400;         // 32x512
  float* dpre2T = ws + 74784;         // 512x32
  float* dpre1T = ws + 91168;         // 512x32  (total ~430 KB)

  k_init<<<86, 256, 0, stream>>>(init_actions, z, flat, xbuf);
  for (int it = 0; it < 8; ++it) {
    k_fwd1<<<8, 128, 0, stream>>>(xbuf, W1, b1, h1, h1T);
    k_fwd2<<<8, 128, 0, stream>>>(h1, W2, b2, W3, h2, dpre2T);
    k_err<<<1, 512, 0, stream>>>(h2, W3, b3, e);
    k_bwd1<<<8, 128, 0, stream>>>(W2, dpre2T, h1T, dpre1T);
    k_bwd2_update<<<1, 224, 0, stream>>>(W1, dpre1T, flat, e, xbuf, out,
                                         it == 7 ? 1 : 0);
  }
}